// LaplacianEmbedding_7894149890111
// MI455X (gfx1250) — compile-verified
//
#include <hip/hip_runtime.h>
#include <hip/hip_bf16.h>

#define NV 8192      // vertexes
#define DEMB 128     // embedding dim
#define SI 132       // padded LDS row stride in floats (conflict-free for ds_load_b64 frags)

typedef __attribute__((ext_vector_type(2))) float v2f;
typedef __attribute__((ext_vector_type(8))) float v8f;

// ---------------------------------------------------------------------------
// Kernel 1: sq[i] = ||w_i||^2 (one wave32 per row), and copy W -> d_out+1.
// ---------------------------------------------------------------------------
__global__ void lap_sq_copy_kernel(const float* __restrict__ W,
                                   float* __restrict__ sq,
                                   float* __restrict__ outw) {
  const int wave = threadIdx.x >> 5;
  const int lane = threadIdx.x & 31;
  const int row  = blockIdx.x * 8 + wave;
  const float4 v = *reinterpret_cast<const float4*>(W + (size_t)row * DEMB + lane * 4);
  float s = v.x * v.x + v.y * v.y + v.z * v.z + v.w * v.w;
  #pragma unroll
  for (int off = 16; off > 0; off >>= 1) s += __shfl_xor(s, off, 32);
  if (lane == 0) sq[row] = s;
  // d_out+1 is only 4B aligned -> scalar stores
  float* o = outw + (size_t)row * DEMB + lane * 4;
  o[0] = v.x; o[1] = v.y; o[2] = v.z; o[3] = v.w;
}

// ---------------------------------------------------------------------------
// Kernel 2: per 128x64 block tile: G = W W^T via V_WMMA_F32_16X16X4_F32,
// then sum A[i,j]*sqrt(max(sq_i+sq_j-2G,0)) -> one partial per block.
// 256 threads = 8 waves (4 M x 2 N); each wave: 32x32 = 2x2 WMMA tiles.
// ---------------------------------------------------------------------------
__global__ void lap_gram_dist_kernel(const float* __restrict__ A,
                                     const float* __restrict__ W,
                                     const float* __restrict__ sq,
                                     float* __restrict__ partial) {
  extern __shared__ float smem[];                // (128+64) rows * SI floats
  const int tid = threadIdx.x;
  const int i0  = blockIdx.x * 128;
  const int j0  = blockIdx.y * 64;

  // ---- stage W rows [i0,i0+128) and [j0,j0+64) into padded LDS ----
  #pragma unroll
  for (int t = tid; t < 192 * 32; t += 256) {
    const int row = t >> 5;                      // 0..191
    const int c4  = (t & 31) << 2;               // 0..124 step 4
    const float4 v = (row < 128)
        ? *reinterpret_cast<const float4*>(W + (size_t)(i0 + row) * DEMB + c4)
        : *reinterpret_cast<const float4*>(W + (size_t)(j0 + row - 128) * DEMB + c4);
    *reinterpret_cast<float4*>(&smem[row * SI + c4]) = v;
  }

  // ---- prefetch this block's A tile (128x64 f32 = 32KB = 512 x 64B lines)
  // so the HBM stream overlaps the WMMA loop instead of serializing after it.
  {
    const float* p0 = A + (size_t)(i0 + (tid >> 2)) * NV + j0 + (tid & 3) * 16;
    const float* p1 = A + (size_t)(i0 + 64 + (tid >> 2)) * NV + j0 + (tid & 3) * 16;
    __builtin_prefetch(p0, 0, 1);
    __builtin_prefetch(p1, 0, 1);
  }
  __syncthreads();

  const int wave = tid >> 5;
  const int lane = tid & 31;
  const int wm = wave >> 1;                      // 0..3  (M)
  const int wn = wave & 1;                       // 0..1  (N)
  const int r  = lane & 15;
  const int g  = lane >> 4;

  // fragment base pointers (ISA 16x4 f32 layout: VGPR0=K{0,2}, VGPR1=K{1,3})
  const float* ap[2];
  const float* bp[2];
  #pragma unroll
  for (int t = 0; t < 2; ++t) {
    ap[t] = &smem[(wm * 32 + t * 16 + r) * SI + 2 * g];
    bp[t] = &smem[(128 + wn * 32 + t * 16 + r) * SI + 2 * g];
  }

  v8f acc[2][2];
  #pragma unroll
  for (int mt = 0; mt < 2; ++mt)
    #pragma unroll
    for (int nt = 0; nt < 2; ++nt) acc[mt][nt] = (v8f){};

  #pragma unroll 8
  for (int k = 0; k < DEMB; k += 4) {
    v2f a[2], b[2];
    #pragma unroll
    for (int t = 0; t < 2; ++t) {
      a[t] = *reinterpret_cast<const v2f*>(ap[t] + k);   // ds_load_2addr_b64
      b[t] = *reinterpret_cast<const v2f*>(bp[t] + k);
    }
    #pragma unroll
    for (int mt = 0; mt < 2; ++mt)
      #pragma unroll
      for (int nt = 0; nt < 2; ++nt)
        acc[mt][nt] = __builtin_amdgcn_wmma_f32_16x16x4_f32(
            false, a[mt], false, b[nt], (short)0, acc[mt][nt], false, false);
  }

  // ---- epilogue: C layout -> M = 8*g + v, N = r ----
  // A is streamed exactly once (256 MB): non-temporal loads keep W/sq hot in L2.
  float lsum = 0.0f;
  #pragma unroll
  for (int mt = 0; mt < 2; ++mt) {
    float sqi[8];
    const int ibase = i0 + wm * 32 + mt * 16 + g * 8;
    #pragma unroll
    for (int v = 0; v < 8; ++v) sqi[v] = sq[ibase + v];
    #pragma unroll
    for (int nt = 0; nt < 2; ++nt) {
      const int j = j0 + wn * 32 + nt * 16 + r;
      const float sqj = sq[j];
      #pragma unroll
      for (int v = 0; v < 8; ++v) {
        const int i = ibase + v;
        float d2 = sqi[v] + sqj - 2.0f * acc[mt][nt][v];
        d2 = fmaxf(d2, 0.0f);
        const float a_ij = __builtin_nontemporal_load(&A[(size_t)i * NV + j]);
        lsum = fmaf(a_ij, __builtin_sqrtf(d2), lsum);
      }
    }
  }

  // ---- block reduction (deterministic order) ----
  #pragma unroll
  for (int off = 16; off > 0; off >>= 1) lsum += __shfl_xor(lsum, off, 32);
  __syncthreads();                               // smem staging no longer needed
  if (lane == 0) smem[wave] = lsum;
  __syncthreads();
  if (tid == 0) {
    float s = 0.0f;
    #pragma unroll
    for (int w = 0; w < 8; ++w) s += smem[w];
    partial[blockIdx.y * gridDim.x + blockIdx.x] = s;
  }
}

// ---------------------------------------------------------------------------
// Kernel 3: deterministic fixed-order reduction of 8192 block partials.
// ---------------------------------------------------------------------------
__global__ void lap_reduce_kernel(const float* __restrict__ partial,
                                  float* __restrict__ out) {
  __shared__ float red[256];
  float s = 0.0f;
  for (int idx = threadIdx.x; idx < 8192; idx += 256) s += partial[idx];
  red[threadIdx.x] = s;
  __syncthreads();
  for (int off = 128; off > 0; off >>= 1) {
    if (threadIdx.x < off) red[threadIdx.x] += red[threadIdx.x + off];
    __syncthreads();
  }
  if (threadIdx.x == 0) out[0] = red[0];
}

extern "C" void kernel_launch(void* const* d_in, const int* in_sizes, int n_in,
                              void* d_out, int out_size, void* d_ws, size_t ws_size,
                              hipStream_t stream) {
  const float* A = (const float*)d_in[0];   // [N,N] f32
  const float* W = (const float*)d_in[1];   // [N,D] f32
  float* out = (float*)d_out;               // [1 + N*D] f32
  float* sq = (float*)d_ws;                 // N floats
  float* partial = sq + NV;                 // 8192 floats

  lap_sq_copy_kernel<<<NV / 8, 256, 0, stream>>>(W, sq, out + 1);

  dim3 grid(NV / 128, NV / 64);             // 64 x 128 = 8192 blocks
  const size_t lds_bytes = (size_t)(128 + 64) * SI * sizeof(float); // ~99 KB
  lap_gram_dist_kernel<<<grid, 256, lds_bytes, stream>>>(A, W, sq, partial);

  lap_reduce_kernel<<<1, 256, 0, stream>>>(partial, out);
}